// Model_88416196755814
// MI455X (gfx1250) — compile-verified
//
#include <hip/hip_runtime.h>
#include <hip/hip_bf16.h>

// out[t] = sum_n softmax(w)[n] * x[n,t]   (top_k with k==n is a permutation,
// which commutes with the weighted sum -> plain softmax-weighted GEMV).
//
// Bandwidth-bound: 819 MB of x @ 23.3 TB/s ~= 35 us floor.
// V_WMMA_F32_16X16X4_F32 is the MAC engine (exact f32): p-chunk in row 0 of A,
// 4x16 x-tile in B, D row 0 accumulates 16 output columns.
// Inner loop is branch-free (cndmask select instead of exec-masked loads) so
// the compiler can unroll/pipeline; x loads are non-temporal (819MB > 192MB L2).

typedef float v2f __attribute__((ext_vector_type(2)));
typedef float v8f __attribute__((ext_vector_type(8)));

#define ROWS_PER_CHUNK 512

// ---------------------------------------------------------------------------
// Kernel 1: softmax statistics over w: m = max(w), inv_s = 1 / sum(exp(w-m))
// Single block of 1024 threads; fixed-order tree reduction (deterministic).
// ---------------------------------------------------------------------------
__global__ void softmax_stats_kernel(const float* __restrict__ w, int n,
                                     float* __restrict__ stats) {
    __shared__ float red[1024];
    const int tid = threadIdx.x;

    float lmax = -3.402823466e38f;
    for (int i = tid; i < n; i += 1024) lmax = fmaxf(lmax, w[i]);
    red[tid] = lmax;
    __syncthreads();
    for (int s = 512; s > 0; s >>= 1) {
        if (tid < s) red[tid] = fmaxf(red[tid], red[tid + s]);
        __syncthreads();
    }
    const float m = red[0];
    __syncthreads();

    float lsum = 0.f;
    for (int i = tid; i < n; i += 1024) lsum += __expf(w[i] - m);
    red[tid] = lsum;
    __syncthreads();
    for (int s = 512; s > 0; s >>= 1) {
        if (tid < s) red[tid] += red[tid + s];
        __syncthreads();
    }
    if (tid == 0) { stats[0] = m; stats[1] = 1.0f / red[0]; }
}

// ---------------------------------------------------------------------------
// Kernel 1b: p[i] = exp(w[i] - m)   (unnormalized softmax; 1/sum applied at end)
// ---------------------------------------------------------------------------
__global__ void compute_p_kernel(const float* __restrict__ w,
                                 const float* __restrict__ stats,
                                 float* __restrict__ p, int n) {
    const int i = blockIdx.x * blockDim.x + threadIdx.x;
    if (i < n) p[i] = __expf(w[i] - stats[0]);
}

// ---------------------------------------------------------------------------
// Kernel 2: WMMA GEMV.  Each wave owns one 16-column tile of one row-chunk.
//
// A (16x4 f32, 2 VGPRs): lanes 0-15 hold M=0-15 with VGPR0=K0,VGPR1=K1;
//                        lanes 16-31 hold M=0-15 with VGPR0=K2,VGPR1=K3.
//   -> only lanes 0 and 16 (M==0) carry the 4 softmax probs; rest zeroed
//      via branch-free select.
// B (4x16 f32, 2 VGPRs): VGPR0: lanes0-15 = row K0, lanes16-31 = row K2;
//                        VGPR1: lanes0-15 = row K1, lanes16-31 = row K3.
// D (16x16 f32, 8 VGPRs): row M=0 lives in acc[0], lanes 0-15 (N = lane).
// ---------------------------------------------------------------------------
__global__ void wmma_gemv_kernel(const float* __restrict__ x,
                                 const float* __restrict__ p,
                                 float* __restrict__ partial,
                                 int n, int t) {
    const int lane = threadIdx.x & 31;
    const int wave = threadIdx.x >> 5;
    const int half = lane >> 4;        // 0: K0/K1 side, 1: K2/K3 side
    const int nl   = lane & 15;        // column within the 16-wide tile
    const int col0 = (blockIdx.x * 8 + wave) * 16;
    if (col0 >= t) return;             // wave-uniform: EXEC stays full for WMMA

    // Clamp OOB columns: B junk only pollutes D columns we never store.
    int col = col0 + nl;
    if (col >= t) col = t - 1;

    const int  r0    = blockIdx.y * ROWS_PER_CHUNK;
    const int  rEnd  = min(r0 + ROWS_PER_CHUNK, n);
    const bool pLane = (nl == 0);      // lanes 0 and 16 carry A row 0

    // Row pointers for this half-wave's two K rows; advance by 4 rows/iter.
    const int k0b = r0 + 2 * half;
    const float* __restrict__ px0 = x + (size_t)k0b * (size_t)t + col;
    const float* __restrict__ px1 = px0 + (size_t)t;
    const float* __restrict__ pp  = p + k0b;           // 8B-aligned (k0b even)
    const size_t step = (size_t)4 * (size_t)t;

    v8f acc = {0.f, 0.f, 0.f, 0.f, 0.f, 0.f, 0.f, 0.f};

    const int iters = (rEnd - r0) >> 2;
#pragma unroll 4
    for (int i = 0; i < iters; ++i) {
        v2f b;
        b.x = __builtin_nontemporal_load(px0);          // x[K0|K2][col]
        b.y = __builtin_nontemporal_load(px1);          // x[K1|K3][col]
        v2f pv = *(const v2f*)pp;                       // broadcast within half-wave
        v2f a;
        a.x = pLane ? pv.x : 0.f;                       // branch-free A row 0
        a.y = pLane ? pv.y : 0.f;
        acc = __builtin_amdgcn_wmma_f32_16x16x4_f32(
            /*neg_a=*/false, a, /*neg_b=*/false, b,
            /*c_mod=*/(short)0, acc, /*reuse_a=*/false, /*reuse_b=*/false);
        px0 += step; px1 += step; pp += 4;
    }

    const int r = r0 + (iters << 2);
    if (r < rEnd) {                    // ragged tail (< 4 rows): zero-pad K
        const int rem = rEnd - r;
        const int k0 = 2 * half, k1 = 2 * half + 1;
        v2f a; a.x = 0.f; a.y = 0.f;
        v2f b; b.x = 0.f; b.y = 0.f;
        if (k0 < rem) {
            b.x = px0[0];
            if (pLane) a.x = pp[0];
        }
        if (k1 < rem) {
            b.y = px1[0];
            if (pLane) a.y = pp[1];
        }
        acc = __builtin_amdgcn_wmma_f32_16x16x4_f32(
            false, a, false, b, (short)0, acc, false, false);
    }

    // D row M=0 = acc[0] in lanes 0-15.
    if (lane < 16 && (col0 + lane) < t) {
        partial[(size_t)blockIdx.y * (size_t)t + (col0 + lane)] = acc[0];
    }
}

// ---------------------------------------------------------------------------
// Kernel 3: reduce partials over row-chunks and apply 1/sum(exp).
// Fully overwrites d_out each call (deterministic, poison-safe).
// ---------------------------------------------------------------------------
__global__ void finalize_kernel(const float* __restrict__ partial,
                                const float* __restrict__ stats,
                                float* __restrict__ out, int t, int numChunks) {
    const int j = blockIdx.x * blockDim.x + threadIdx.x;
    if (j >= t) return;
    float s = 0.f;
    for (int c = 0; c < numChunks; ++c) s += partial[(size_t)c * (size_t)t + j];
    out[j] = s * stats[1];
}

// ---------------------------------------------------------------------------
extern "C" void kernel_launch(void* const* d_in, const int* in_sizes, int n_in,
                              void* d_out, int out_size, void* d_ws, size_t ws_size,
                              hipStream_t stream) {
    const float* x = (const float*)d_in[0];   // (n, t) f32
    const float* w = (const float*)d_in[1];   // (n,)   f32
    // d_in[2] is k == n (full top-k == permutation), not needed.

    const int n = in_sizes[1];
    const int t = in_sizes[0] / n;            // 4096

    float* ws_f    = (float*)d_ws;
    float* stats   = ws_f;                    // [0]=max, [1]=1/sum
    float* p       = ws_f + 16;               // n floats (unnormalized softmax)
    float* partial = p + ((n + 15) & ~15);    // numChunks * t floats

    const int numChunks = (n + ROWS_PER_CHUNK - 1) / ROWS_PER_CHUNK;

    softmax_stats_kernel<<<1, 1024, 0, stream>>>(w, n, stats);
    compute_p_kernel<<<(n + 255) / 256, 256, 0, stream>>>(w, stats, p, n);

    const int colGroups = (t + 127) / 128;    // 8 waves/block * 16 cols/wave
    dim3 grid(colGroups, numChunks);
    wmma_gemv_kernel<<<grid, 256, 0, stream>>>(x, p, partial, n, t);

    finalize_kernel<<<(t + 255) / 256, 256, 0, stream>>>(
        partial, stats, (float*)d_out, t, numChunks);
}